// GridExp_14233521619392
// MI455X (gfx1250) — compile-verified
//
#include <hip/hip_runtime.h>

// Scaling-and-squaring of a stationary velocity field, circular (dft) bounds.
// velocity: (2,128,128,128,3) f32, channel order (z,y,x).
// v = vel/2^8 ; 8x: v <- v + trilerp_wrap(v, id+v) ; out = id + v.
//
// MI455X design: direct-gather kernel. The 50MB field is L2-resident (192MB
// global L2), so the 8-corner gathers are cache hits; we maximize memory-level
// parallelism (8 independent b96 corner loads per thread, 8 wave32 waves per
// 256-thread block) instead of LDS halo tiling (the halo is unbounded for
// Gaussian velocities, and LDS staging adds a round trip that L2 already
// covers). Wrap is & 127 (dims are powers of two; valid for negative ints in
// two's complement). The 1/2^8 scale folds into sweep 0 (interp is linear);
// the identity-grid add folds into sweep 7. global_prefetch_b8 warms L2 ahead
// of the streaming self-read.

#define MASK   127
#define VOXELS_PER_BATCH (128 * 128 * 128)      // 2,097,152
#define ELEMS_PER_BATCH  (VOXELS_PER_BATCH * 3) // 6,291,456

template <bool SCALE_FIRST, bool ADD_GRID>
__global__ __launch_bounds__(256) void gridexp_iter_kernel(
    const float* __restrict__ src,   // v_k field (raw velocity when SCALE_FIRST)
    float*       __restrict__ dst)   // v_{k+1} (id + v_K when ADD_GRID)
{
    // Exact launch: no bounds check, no execz path.
    const unsigned tid = blockIdx.x * 256u + threadIdx.x;

    const int x = (int)( tid        & MASK);
    const int y = (int)((tid >> 7)  & MASK);
    const int z = (int)((tid >> 14) & MASK);
    const int b = (int)( tid >> 21);

    const float* __restrict__ sb = src + (long long)b * ELEMS_PER_BATCH;
    float*       __restrict__ db = dst + (long long)b * ELEMS_PER_BATCH;

    const int self = ((((z << 7) | y) << 7) | x) * 3;

    // gfx1250: warm L2 ahead of the streaming self-read (global_prefetch_b8).
    __builtin_prefetch(sb + self + 3 * 4096, 0, 1);

    constexpr float scale = SCALE_FIRST ? (1.0f / 256.0f) : 1.0f;

    // Self displacement (scaled) -> sampling coordinate phi = id + v.
    const float vz = scale * sb[self + 0];
    const float vy = scale * sb[self + 1];
    const float vx = scale * sb[self + 2];

    const float pz = (float)z + vz;
    const float py = (float)y + vy;
    const float px = (float)x + vx;

    const float fz = floorf(pz), fy = floorf(py), fx = floorf(px);
    const float tz = pz - fz,    ty = py - fy,    tx = px - fx;

    const int iz0 = ((int)fz) & MASK, iz1 = (iz0 + 1) & MASK;
    const int iy0 = ((int)fy) & MASK, iy1 = (iy0 + 1) & MASK;
    const int ix0 = ((int)fx) & MASK, ix1 = (ix0 + 1) & MASK;

    const int   zi[2] = { iz0, iz1 };   const float wz[2] = { 1.0f - tz, tz };
    const int   yi[2] = { iy0, iy1 };   const float wy[2] = { 1.0f - ty, ty };
    const int   xi[2] = { ix0, ix1 };   const float wx[2] = { 1.0f - tx, tx };

    // 8-corner trilinear gather on the *unscaled* field; scale applied once at
    // the end (interp(s*v) == s*interp(v)). All 8 b96 loads are independent ->
    // deep MLP; mostly L2 hits on MI455X.
    float az = 0.0f, ay = 0.0f, ax = 0.0f;
#pragma unroll
    for (int a = 0; a < 2; ++a) {
#pragma unroll
        for (int bb = 0; bb < 2; ++bb) {
            const int zy = ((zi[a] << 7) | yi[bb]) << 7;
            const float wzy = wz[a] * wy[bb];
#pragma unroll
            for (int c = 0; c < 2; ++c) {
                const int off = (zy | xi[c]) * 3;
                const float w = wzy * wx[c];
                az = fmaf(w, sb[off + 0], az);
                ay = fmaf(w, sb[off + 1], ay);
                ax = fmaf(w, sb[off + 2], ax);
            }
        }
    }

    float oz, oy, ox;
    if constexpr (SCALE_FIRST) {
        oz = vz + scale * az;
        oy = vy + scale * ay;
        ox = vx + scale * ax;
    } else {
        oz = vz + az;
        oy = vy + ay;
        ox = vx + ax;
    }
    if constexpr (ADD_GRID) { oz += (float)z; oy += (float)y; ox += (float)x; }

    db[self + 0] = oz;
    db[self + 1] = oy;
    db[self + 2] = ox;
}

extern "C" void kernel_launch(void* const* d_in, const int* in_sizes, int n_in,
                              void* d_out, int out_size, void* d_ws, size_t ws_size,
                              hipStream_t stream)
{
    (void)n_in; (void)out_size; (void)ws_size;

    const float* vel = (const float*)d_in[0];
    float*       out = (float*)d_out;
    float*       ws  = (float*)d_ws;     // one 50.3MB ping-pong buffer

    const unsigned total_voxels = (unsigned)(in_sizes[0] / 3); // 4,194,304
    const dim3 block(256);
    const dim3 grid(total_voxels / 256u);  // exact multiple: 16384 blocks

    // 8 sweeps. Ping-pong ws <-> d_out; only kernel launches (graph-safe).
    // i0: vel -> ws   (scale 1/2^8 folded in)
    // i1..i6: alternate ws/out
    // i7: ws  -> out  (+ identity grid)
    gridexp_iter_kernel<true,  false><<<grid, block, 0, stream>>>(vel, ws);
    gridexp_iter_kernel<false, false><<<grid, block, 0, stream>>>(ws,  out);
    gridexp_iter_kernel<false, false><<<grid, block, 0, stream>>>(out, ws);
    gridexp_iter_kernel<false, false><<<grid, block, 0, stream>>>(ws,  out);
    gridexp_iter_kernel<false, false><<<grid, block, 0, stream>>>(out, ws);
    gridexp_iter_kernel<false, false><<<grid, block, 0, stream>>>(ws,  out);
    gridexp_iter_kernel<false, false><<<grid, block, 0, stream>>>(out, ws);
    gridexp_iter_kernel<false, true ><<<grid, block, 0, stream>>>(ws,  out);
}